// Loss_19189913878893
// MI455X (gfx1250) — compile-verified
//
#include <hip/hip_runtime.h>

// ---------------------------------------------------------------------------
// Problem constants (fixed by the reference: B=32, N=131072, alpha=beta=gamma=1)
// ---------------------------------------------------------------------------
namespace {
constexpr int B_            = 32;
constexpr int N_            = 131072;
constexpr int PTS_PER_WAVE  = 1024;                 // points per wave sub-chunk
constexpr int SUBCH         = N_ / PTS_PER_WAVE;    // 128 sub-chunks per batch
constexpr int BLOCK         = 256;                  // 8 wave32's per block
constexpr int WAVES_PER_BLK = BLOCK / 32;
constexpr int TOT_WAVES     = B_ * SUBCH;           // 4096 waves
constexpr int GRID_MAIN     = TOT_WAVES / WAVES_PER_BLK; // 512 blocks
constexpr int NACC          = 10;                   // upper triangle of 4x4 moment matrix
}

typedef __attribute__((ext_vector_type(2))) float v2f;
typedef __attribute__((ext_vector_type(8))) float v8f;

__device__ __forceinline__ unsigned ballot32(bool p) {
#if __has_builtin(__builtin_amdgcn_ballot_w32)
  return __builtin_amdgcn_ballot_w32(p);
#else
  return (unsigned)__ballot(p ? 1 : 0);   // wave32: mask lives in low 32 bits
#endif
}

// ---------------------------------------------------------------------------
// k_zero: zero the per-batch moment accumulators (ws is poisoned by harness)
// ---------------------------------------------------------------------------
__global__ void k_zero(float* __restrict__ accum) {
  int t = threadIdx.x;
  if (t < B_ * NACC) accum[t] = 0.0f;
}

// ---------------------------------------------------------------------------
// Pass 1: per-(batch, sub-chunk) masked-point counts. Streams all of velo
// through L2 (192 MB; the 64 MiB tensor stays resident for pass 2).
// ---------------------------------------------------------------------------
__global__ __launch_bounds__(BLOCK) void k_count(const float4* __restrict__ velo,
                                                 int* __restrict__ counts) {
  const int wid  = (blockIdx.x * BLOCK + threadIdx.x) >> 5;
  const int lane = threadIdx.x & 31;
  const int b    = wid / SUBCH;
  const int sc   = wid % SUBCH;
  const long base = (long)b * N_ + (long)sc * PTS_PER_WAVE;

  int cnt = 0;
#pragma unroll 4
  for (int it = 0; it < PTS_PER_WAVE / 32; ++it) {
    __builtin_prefetch((const void*)(velo + base + (it + 8) * 32 + lane), 0, 1);
    float4 p = velo[base + it * 32 + lane];
    cnt += (p.w > 0.0f) ? 1 : 0;
  }
#pragma unroll
  for (int s = 16; s > 0; s >>= 1) cnt += __shfl_xor(cnt, s, 32);
  if (lane == 0) counts[wid] = cnt;
}

// ---------------------------------------------------------------------------
// Scan: exclusive prefix of sub-chunk counts per batch + min_pts over batches.
// One wave32; lane b owns batch b (128 serial adds — negligible).
// ---------------------------------------------------------------------------
__global__ void k_scan(const int* __restrict__ counts,
                       int* __restrict__ offs,
                       int* __restrict__ minp) {
  const int b = threadIdx.x;              // 0..31 == batch
  const int base = b * SUBCH;
  int off = 0;
  for (int i = 0; i < SUBCH; ++i) {
    offs[base + i] = off;
    off += counts[base + i];
  }
  int total = off;
#pragma unroll
  for (int s = 16; s > 0; s >>= 1) total = min(total, __shfl_xor(total, s, 32));
  if (b == 0) minp[0] = total;
}

// ---------------------------------------------------------------------------
// Pass 2 (main): stream points (L2-hot), compute per-lane masked rank via
// ballot/popc, select rank < min_pts, accumulate the 10 symmetric moments
//   S = sum over selected points of p_h p_h^T,  p_h = (x,y,z,1)
// order: [xx xy xz x | yy yz y | zz z | n]
// ---------------------------------------------------------------------------
__global__ __launch_bounds__(BLOCK) void k_main(const float4* __restrict__ velo,
                                                const int* __restrict__ offs,
                                                const int* __restrict__ minp,
                                                float* __restrict__ accum) {
  const int wid  = (blockIdx.x * BLOCK + threadIdx.x) >> 5;
  const int lane = threadIdx.x & 31;
  const int b    = wid / SUBCH;
  const int sc   = wid % SUBCH;
  const long base = (long)b * N_ + (long)sc * PTS_PER_WAVE;
  const unsigned lt_mask = (1u << lane) - 1u;

  int off = offs[wid];
  const int mp = minp[0];
  const bool active = (off < mp);

  float acc[NACC];
#pragma unroll
  for (int j = 0; j < NACC; ++j) acc[j] = 0.0f;

  if (active) {
#pragma unroll 2
    for (int it = 0; it < PTS_PER_WAVE / 32; ++it) {
      __builtin_prefetch((const void*)(velo + base + (it + 8) * 32 + lane), 0, 1);
      float4 p = velo[base + it * 32 + lane];
      const bool m = (p.w > 0.0f);
      const unsigned bal = ballot32(m);
      const int pre = __popc(bal & lt_mask);
      const float w = (m && (off + pre) < mp) ? 1.0f : 0.0f;
      const float x = p.x * w, y = p.y * w, z = p.z * w;
      acc[0] += x * x; acc[1] += x * y; acc[2] += x * z; acc[3] += x;
      acc[4] += y * y; acc[5] += y * z; acc[6] += y;
      acc[7] += z * z; acc[8] += z;
      acc[9] += w;
      off += __popc(bal);
      if (off >= mp) break;              // uniform across the wave
    }
#pragma unroll
    for (int s = 16; s > 0; s >>= 1) {
#pragma unroll
      for (int j = 0; j < NACC; ++j) acc[j] += __shfl_xor(acc[j], s, 32);
    }
    if (lane == 0) {
#pragma unroll
      for (int j = 0; j < NACC; ++j) atomicAdd(&accum[b * NACC + j], acc[j]);
    }
  }
}

// ---------------------------------------------------------------------------
// Finalize: build SE(3) matrices, chain them with V_WMMA_F32_16X16X4_F32
// (true 4x4x4 matrix products, rows/cols >= 4 zero-padded), then the tiny
// quadratic-form loss assembly. One wave32 (EXEC all ones for WMMA).
// ---------------------------------------------------------------------------
__device__ __forceinline__ void build_T(const float r[3], const float t[3], float* T) {
  // R = Rz @ Ry @ Rx (reference euler convention)
  const float cx = cosf(r[0]), sx = sinf(r[0]);
  const float cy = cosf(r[1]), sy = sinf(r[1]);
  const float cz = cosf(r[2]), sz = sinf(r[2]);
  T[0]  = cz * cy; T[1]  = cz * sy * sx - sz * cx; T[2]  = cz * sy * cx + sz * sx; T[3]  = t[0];
  T[4]  = sz * cy; T[5]  = sz * sy * sx + cz * cx; T[6]  = sz * sy * cx - cz * sx; T[7]  = t[1];
  T[8]  = -sy;     T[9]  = cy * sx;                T[10] = cy * cx;                T[11] = t[2];
  T[12] = 0.0f;    T[13] = 0.0f;                   T[14] = 0.0f;                   T[15] = 1.0f;
}

__global__ void k_final(const float* __restrict__ rot_pred,  const float* __restrict__ trans_pred,
                        const float* __restrict__ rot_gt,    const float* __restrict__ trans_gt,
                        const float* __restrict__ Tg,
                        const float* __restrict__ rot_mean,  const float* __restrict__ rot_std,
                        const float* __restrict__ trans_mean,const float* __restrict__ trans_std,
                        const float* __restrict__ accum,     const int* __restrict__ minp,
                        float* __restrict__ out) {
  __shared__ float Terr[B_][16];
  __shared__ float Tfix[B_][16];
  __shared__ float Tin [B_][16];
  __shared__ float Trec[B_][16];
  __shared__ float Mtmp[16];

  const int b = threadIdx.x;             // lane == batch, 32 lanes total

  // --- per-lane scalar setup -------------------------------------------------
  float re[3], te[3], rp[3], tp[3];
  float sr = 0.0f, st = 0.0f;            // mse partials (raw pred vs gt)
#pragma unroll
  for (int i = 0; i < 3; ++i) {
    const float rpi = rot_pred[b * 3 + i],   rgi = rot_gt[b * 3 + i];
    const float tpi = trans_pred[b * 3 + i], tgi = trans_gt[b * 3 + i];
    rp[i] = rpi * rot_std[i]   + rot_mean[i];
    tp[i] = tpi * trans_std[i] + trans_mean[i];
    re[i] = rgi * rot_std[i]   + rot_mean[i];
    te[i] = tgi * trans_std[i] + trans_mean[i];
    float d = rpi - rgi; sr += d * d;
    d = tpi - tgi;       st += d * d;
  }
  build_T(re, te, &Terr[b][0]);
  float Tp[16];
  build_T(rp, tp, Tp);
  // T_fix = inv(T_p) = [R^T | -R^T t]
  float* F = &Tfix[b][0];
#pragma unroll
  for (int r = 0; r < 3; ++r) {
#pragma unroll
    for (int c = 0; c < 3; ++c) F[r * 4 + c] = Tp[c * 4 + r];
    F[r * 4 + 3] = -(Tp[0 * 4 + r] * Tp[3] + Tp[1 * 4 + r] * Tp[7] + Tp[2 * 4 + r] * Tp[11]);
  }
  F[12] = 0.0f; F[13] = 0.0f; F[14] = 0.0f; F[15] = 1.0f;
#pragma unroll
  for (int j = 0; j < 16; ++j) Tin[b][j] = Tg[b * 16 + j];
  __syncthreads();

  // --- WMMA chain: Trec[bb] = Tfix[bb] @ (Terr[bb] @ Tin[bb]) ---------------
  // A (16x4 f32): lane l -> M = l&15, reg v -> K = v + 2*(l>=16)  (ISA 7.12.2)
  // B (4x16 f32): lane l -> N = l&15, reg v -> K = v + 2*(l>=16)
  // D (16x16 f32): reg r, lanes 0-15 -> D[r][lane]
  const int half = b >> 4;
  const int idx  = b & 15;
  const int k0   = 2 * half;
  const bool row4 = (idx < 4);

  for (int bb = 0; bb < B_; ++bb) {
    v8f c0 = {};
    v2f a, bv;
    const float* L = &Terr[bb][0];
    const float* R = &Tin[bb][0];
    a[0]  = row4 ? L[idx * 4 + k0]     : 0.0f;
    a[1]  = row4 ? L[idx * 4 + k0 + 1] : 0.0f;
    bv[0] = row4 ? R[(k0)     * 4 + idx] : 0.0f;
    bv[1] = row4 ? R[(k0 + 1) * 4 + idx] : 0.0f;
    v8f d = __builtin_amdgcn_wmma_f32_16x16x4_f32(false, a, false, bv,
                                                  (short)0, c0, false, false);
    if (half == 0 && idx < 4) {
      Mtmp[0 * 4 + idx] = d[0]; Mtmp[1 * 4 + idx] = d[1];
      Mtmp[2 * 4 + idx] = d[2]; Mtmp[3 * 4 + idx] = d[3];
    }
    __syncthreads();

    const float* L2 = &Tfix[bb][0];
    a[0]  = row4 ? L2[idx * 4 + k0]     : 0.0f;
    a[1]  = row4 ? L2[idx * 4 + k0 + 1] : 0.0f;
    bv[0] = row4 ? Mtmp[(k0)     * 4 + idx] : 0.0f;
    bv[1] = row4 ? Mtmp[(k0 + 1) * 4 + idx] : 0.0f;
    v8f d2 = __builtin_amdgcn_wmma_f32_16x16x4_f32(false, a, false, bv,
                                                   (short)0, c0, false, false);
    if (half == 0 && idx < 4) {
      Trec[bb][0 * 4 + idx] = d2[0]; Trec[bb][1 * 4 + idx] = d2[1];
      Trec[bb][2 * 4 + idx] = d2[2]; Trec[bb][3 * 4 + idx] = d2[3];
    }
    __syncthreads();
  }

  // --- loss assembly (lane b handles batch b) -------------------------------
  float Dm[3][4];
#pragma unroll
  for (int r = 0; r < 3; ++r)
#pragma unroll
    for (int c = 0; c < 4; ++c) Dm[r][c] = Trec[b][r * 4 + c] - Tin[b][r * 4 + c];

  float A10[NACC];
#pragma unroll
  for (int j = 0; j < NACC; ++j) A10[j] = accum[b * NACC + j];
  float S[4][4];
  S[0][0] = A10[0]; S[0][1] = A10[1]; S[0][2] = A10[2]; S[0][3] = A10[3];
  S[1][0] = A10[1]; S[1][1] = A10[4]; S[1][2] = A10[5]; S[1][3] = A10[6];
  S[2][0] = A10[2]; S[2][1] = A10[5]; S[2][2] = A10[7]; S[2][3] = A10[8];
  S[3][0] = A10[3]; S[3][1] = A10[6]; S[3][2] = A10[8]; S[3][3] = A10[9];

  const float nf = (float)minp[0];
  const float s4[4] = { S[0][3], S[1][3], S[2][3], S[3][3] };  // sum of p_h

  float cpart = 0.0f;   // sum_i ((c_r - c_o)_i)^2 for this batch
  float quad  = 0.0f;   // sum_i D_i . S . D_i^T  (loss_depth numerator)
#pragma unroll
  for (int i = 0; i < 3; ++i) {
    const float cd = (Dm[i][0] * s4[0] + Dm[i][1] * s4[1] +
                      Dm[i][2] * s4[2] + Dm[i][3] * s4[3]) / nf;
    cpart += cd * cd;
#pragma unroll
    for (int j = 0; j < 4; ++j) {
      const float tj = S[j][0] * Dm[i][0] + S[j][1] * Dm[i][1] +
                       S[j][2] * Dm[i][2] + S[j][3] * Dm[i][3];
      quad += Dm[i][j] * tj;
    }
  }

  // reduce (sr, st, cpart, quad) across the wave
#pragma unroll
  for (int s = 16; s > 0; s >>= 1) {
    sr    += __shfl_xor(sr, s, 32);
    st    += __shfl_xor(st, s, 32);
    cpart += __shfl_xor(cpart, s, 32);
    quad  += __shfl_xor(quad, s, 32);
  }
  if (b == 0) {
    const float loss_mse    = sr / 96.0f + st / 96.0f;     // ALPHA=1, B*3=96
    const float loss_center = cpart / 96.0f;               // BETA=1, mean over B*3
    const float loss_depth  = quad / (128.0f * nf);        // GAMMA=1, B*4*nf
    out[0] = loss_mse + loss_center + loss_depth;
  }
}

// ---------------------------------------------------------------------------
// Launch: inputs in setup_inputs() order:
// 0 rot_pred 1 trans_pred 2 rot_gt 3 trans_gt 4 velo 5 T
// 6 rot_mean 7 rot_std 8 trans_mean 9 trans_std
// Workspace usage ~35 KB: counts[4096] | offs[4096] | minp(+pad 64) | accum[320]
// ---------------------------------------------------------------------------
extern "C" void kernel_launch(void* const* d_in, const int* in_sizes, int n_in,
                              void* d_out, int out_size, void* d_ws, size_t ws_size,
                              hipStream_t stream) {
  const float* rot_pred   = (const float*)d_in[0];
  const float* trans_pred = (const float*)d_in[1];
  const float* rot_gt     = (const float*)d_in[2];
  const float* trans_gt   = (const float*)d_in[3];
  const float* velo       = (const float*)d_in[4];
  const float* Tg         = (const float*)d_in[5];
  const float* rot_mean   = (const float*)d_in[6];
  const float* rot_std    = (const float*)d_in[7];
  const float* trans_mean = (const float*)d_in[8];
  const float* trans_std  = (const float*)d_in[9];

  int*   counts = (int*)d_ws;
  int*   offs   = counts + TOT_WAVES;
  int*   minp   = offs + TOT_WAVES;
  float* accum  = (float*)d_ws + (2 * TOT_WAVES + 64);

  k_zero <<<1, 512, 0, stream>>>(accum);
  k_count<<<GRID_MAIN, BLOCK, 0, stream>>>((const float4*)velo, counts);
  k_scan <<<1, 32, 0, stream>>>(counts, offs, minp);
  k_main <<<GRID_MAIN, BLOCK, 0, stream>>>((const float4*)velo, offs, minp, accum);
  k_final<<<1, 32, 0, stream>>>(rot_pred, trans_pred, rot_gt, trans_gt, Tg,
                                rot_mean, rot_std, trans_mean, trans_std,
                                accum, minp, (float*)d_out);
}